// CensusTransform_1915555414483
// MI455X (gfx1250) — compile-verified
//
#include <hip/hip_runtime.h>
#include <stdint.h>

// Census transform w/ steep sigmoid, MI455X (gfx1250).
// x: (2,3,384,1280) f32 ; out: (2,120,384,1280) f32.
// Store-bandwidth bound: 472MB NT stores ~= 21us floor at 23.3 TB/s.

#define HFPAD 5
#define NOFF  120
#define TTY   16
#define TTX   64
#define SH_H  (TTY + 2*HFPAD)   // 26
#define SH_W  (TTX + 2*HFPAD)   // 74
#define SH_N  (SH_H * SH_W)     // 1924
#define IMH   384
#define IMW   1280
#define NCH   3
#define NBATCH 2

typedef __attribute__((ext_vector_type(4))) float v4f;

// Offset table in the exact ring order of the Python reference, stored as
// LDS deltas (v*SH_W + u). Center delta is HFPAD*SH_W + HFPAD.
struct OffTab { unsigned short d[NOFF]; };
constexpr OffTab make_offs() {
  OffTab t{};
  bool seen[11][11] = {};
  int n = 0;
  for (int i = 1; i <= HFPAD; ++i)
    for (int v = HFPAD - i; v <= HFPAD + i; ++v)
      for (int u = HFPAD - i; u <= HFPAD + i; ++u) {
        if (u == HFPAD && v == HFPAD) continue;
        if (seen[v][u]) continue;
        seen[v][u] = true;
        t.d[n++] = (unsigned short)(v * SH_W + u);
      }
  return t;
}
__constant__ OffTab OFFS = make_offs();

__device__ __forceinline__ int refl(int m, int n) {
  m = (m < 0) ? -m : m;              // reflect (no edge repeat), pad <= 5
  m = (m >= n) ? (2 * n - 2 - m) : m;
  return m;
}

__device__ __forceinline__ float fast_sigmoid_scaled(float d, float n_scale_log2e) {
  // sigmoid(d*1e5) = 1/(1+exp2(d * (-1e5*log2e)))
  float e = __builtin_amdgcn_exp2f(d * n_scale_log2e);
  return __builtin_amdgcn_rcpf(1.0f + e);
}

__global__ __launch_bounds__(256)
void census_sigmoid_kernel(const float* __restrict__ x,
                           const int* __restrict__ attackp,
                           float* __restrict__ out) {
  __shared__ float ch[NCH][SH_N];    // 3 channel halos; ch[0] is reused as mean tile

  const int tid = threadIdx.x;
  const int gx0 = blockIdx.x * TTX;
  const int gy0 = blockIdx.y * TTY;
  const int nb  = blockIdx.z;

  // ---- Stage 1: async-copy 3-channel reflect-padded halo (26x74) into LDS ----
  const float* xbase = x + (size_t)nb * (NCH * IMH * IMW);
  for (int e = tid; e < NCH * SH_N; e += 256) {
    int c  = e / SH_N;
    int r  = e - c * SH_N;
    int hy = r / SH_W;
    int hx = r - hy * SH_W;
    int gy = refl(gy0 + hy - HFPAD, IMH);
    int gx = refl(gx0 + hx - HFPAD, IMW);
    uint32_t goff = (uint32_t)(((c * IMH + gy) * IMW + gx) * 4);
    uint32_t lds  = (uint32_t)(uintptr_t)(&ch[0][0] + e);
    // gfx1250 async global->LDS (GVS mode), tracked with ASYNCcnt
    asm volatile("global_load_async_to_lds_b32 %0, %1, %2"
                 :: "v"(lds), "v"(goff), "s"(xbase)
                 : "memory");
  }
  asm volatile("s_wait_asynccnt 0x0" ::: "memory");
  __syncthreads();

  // ---- Stage 2: channel mean in-place into ch[0] ----
  for (int e = tid; e < SH_N; e += 256) {
    ch[0][e] = (ch[0][e] + ch[1][e] + ch[2][e]) * (1.0f / 3.0f);
  }
  __syncthreads();

  // ---- Stage 3: 120 shifted comparisons, 4 pixels per thread, NT B128 stores ----
  const float* sh = ch[0];
  const int ty   = tid >> 4;          // 0..15
  const int tx   = (tid & 15) * 4;    // 0..60, quad of pixels
  const int base = ty * SH_W + tx;

  const float c0 = sh[base + (HFPAD * SH_W + HFPAD) + 0];
  const float c1 = sh[base + (HFPAD * SH_W + HFPAD) + 1];
  const float c2 = sh[base + (HFPAD * SH_W + HFPAD) + 2];
  const float c3 = sh[base + (HFPAD * SH_W + HFPAD) + 3];

  const int gy = gy0 + ty;
  const int gx = gx0 + tx;
  float* op = out + (((size_t)nb * NOFF) * IMH + gy) * IMW + gx;
  const size_t kstride = (size_t)IMH * IMW;

  const int attack = *attackp;
  const float NSCALE = -100000.0f * 1.44269504088896340736f; // -scale*log2(e)

  if (attack) {
    for (int k = 0; k < NOFF; ++k) {
      const int dlt = OFFS.d[k];
      float s0 = sh[base + dlt + 0];
      float s1 = sh[base + dlt + 1];
      float s2 = sh[base + dlt + 2];
      float s3 = sh[base + dlt + 3];
      v4f r;
      r.x = fast_sigmoid_scaled(s0 - c0, NSCALE);
      r.y = fast_sigmoid_scaled(s1 - c1, NSCALE);
      r.z = fast_sigmoid_scaled(s2 - c2, NSCALE);
      r.w = fast_sigmoid_scaled(s3 - c3, NSCALE);
      __builtin_nontemporal_store(r, (v4f*)op);
      op += kstride;
    }
  } else {
    for (int k = 0; k < NOFF; ++k) {
      const int dlt = OFFS.d[k];
      v4f r;
      r.x = (sh[base + dlt + 0] >= c0) ? 1.0f : 0.0f;
      r.y = (sh[base + dlt + 1] >= c1) ? 1.0f : 0.0f;
      r.z = (sh[base + dlt + 2] >= c2) ? 1.0f : 0.0f;
      r.w = (sh[base + dlt + 3] >= c3) ? 1.0f : 0.0f;
      __builtin_nontemporal_store(r, (v4f*)op);
      op += kstride;
    }
  }
}

extern "C" void kernel_launch(void* const* d_in, const int* in_sizes, int n_in,
                              void* d_out, int out_size, void* d_ws, size_t ws_size,
                              hipStream_t stream) {
  (void)in_sizes; (void)n_in; (void)d_ws; (void)ws_size; (void)out_size;
  const float* x      = (const float*)d_in[0];
  const int*   attack = (const int*)d_in[1];
  float*       out    = (float*)d_out;

  dim3 grid(IMW / TTX, IMH / TTY, NBATCH); // 20 x 24 x 2 = 960 blocks
  census_sigmoid_kernel<<<grid, 256, 0, stream>>>(x, attack, out);
}